// torchMarginLoss_8890582302787
// MI455X (gfx1250) — compile-verified
//
#include <hip/hip_runtime.h>

// Margin ranking loss (reduction='mean' over negatives per rank j, summed over
// j and batch) for B segments of N=50 hypotheses.
//
// Strategy (MI455X / gfx1250, wave32):
//  - One 32-lane wave per block; each wave handles SEGS_PER_BLOCK segments.
//  - Per segment: gather ranked scores into LDS (padded to 64 with zeros),
//    then build the pairwise matrix  D[j,k] = margin - r[j] + r[k]  as a
//    rank-2 update on the matrix pipe:  V_WMMA_F32_16X16X4_F32 with
//      A[:,0] = -r_j, A[:,1] = 1, B[0,:] = 1, B[1,:] = r_k, C = margin.
//    4x4 tiles of 16x16 cover the padded 64x64 space (16 WMMAs / segment).
//  - relu + (k>j) mask + 1/(N-1-j) weight applied per lane on the D fragment
//    using the ISA C/D layout, accumulated, then shfl-reduced per wave.
//  - Stage 2 kernel reduces per-block partials in a fixed tree order
//    (deterministic: no float atomics).

typedef __attribute__((ext_vector_type(2))) float v2f;
typedef __attribute__((ext_vector_type(8))) float v8f;

#define MARGIN 0.1f
#define NBEST 50
#define NPAD 64           // 4 tiles of 16
#define NTILE 4
#define SEGS_PER_BLOCK 8

__global__ __launch_bounds__(32)
void margin_loss_wmma_kernel(const float* __restrict__ scores,
                             const int*   __restrict__ werRank,
                             float*       __restrict__ partials,
                             int B)
{
    __shared__ float r[NPAD];

    const int  lane  = threadIdx.x;      // 0..31, all lanes active (EXEC all 1s)
    const int  loIdx = lane & 15;
    const bool hi    = lane >= 16;

    float blockAcc = 0.0f;
    const int seg0 = blockIdx.x * SEGS_PER_BLOCK;

    for (int s = 0; s < SEGS_PER_BLOCK; ++s) {
        const int seg = seg0 + s;        // uniform across the wave
        if (seg >= B) break;

        // Gather ranked scores into LDS; pad [NBEST, NPAD) with 0.
        const long base = (long)seg * NBEST;
        for (int i = lane; i < NPAD; i += 32) {
            float v = 0.0f;
            if (i < NBEST) v = scores[base + (long)werRank[base + i]];
            r[i] = v;
        }
        __syncthreads();

        float acc = 0.0f;

        for (int jt = 0; jt < NTILE; ++jt) {
            // A fragment (16x4 f32): lanes 0-15 hold K=0 (VGPR0) and K=1
            // (VGPR1) for row M=lane; lanes 16-31 hold K=2/K=3 -> zero.
            v2f a;
            a.x = hi ? 0.0f : -r[jt * 16 + loIdx];   // K=0 column: -r_j
            a.y = hi ? 0.0f : 1.0f;                  // K=1 column: 1

            // Row weights 1/(N-1-j) for this jt, per D-fragment VGPR slot.
            float wrow[8];
#pragma unroll
            for (int v = 0; v < 8; ++v) {
                const int j = jt * 16 + v + (hi ? 8 : 0);
                wrow[v] = (j < NBEST - 1) ? (1.0f / (float)(NBEST - 1 - j)) : 0.0f;
            }

            for (int kt = 0; kt < NTILE; ++kt) {
                // B fragment (4x16 f32): lanes 0-15 hold rows K=0 (VGPR0) and
                // K=1 (VGPR1) at column N=lane; lanes 16-31 (K=2/3) -> zero.
                v2f b;
                b.x = hi ? 0.0f : 1.0f;                  // row K=0: ones
                b.y = hi ? 0.0f : r[kt * 16 + loIdx];    // row K=1: r_k

                v8f c = {MARGIN, MARGIN, MARGIN, MARGIN,
                         MARGIN, MARGIN, MARGIN, MARGIN};

                // D[j,k] = margin - r[j] + r[k]
                v8f d = __builtin_amdgcn_wmma_f32_16x16x4_f32(
                    /*neg_a=*/false, a, /*neg_b=*/false, b,
                    /*c_mod=*/(short)0, c, /*reuse_a=*/false, /*reuse_b=*/false);

                const int k = kt * 16 + loIdx;
#pragma unroll
                for (int v = 0; v < 8; ++v) {
                    const int j = jt * 16 + v + (hi ? 8 : 0);
                    const float w = (k > j && k < NBEST) ? wrow[v] : 0.0f;
                    acc += fmaxf(d[v], 0.0f) * w;
                }
            }
        }

        __syncthreads();   // protect LDS before next segment overwrites it
        blockAcc += acc;
    }

    // Wave32 reduction
    for (int off = 16; off > 0; off >>= 1)
        blockAcc += __shfl_down(blockAcc, off, 32);

    if (lane == 0) partials[blockIdx.x] = blockAcc;
}

__global__ __launch_bounds__(256)
void reduce_partials_kernel(const float* __restrict__ partials, int n,
                            float* __restrict__ out)
{
    __shared__ float sm[256];
    const int t = threadIdx.x;
    float s = 0.0f;
    for (int i = t; i < n; i += 256) s += partials[i];
    sm[t] = s;
    __syncthreads();
    for (int off = 128; off > 0; off >>= 1) {
        if (t < off) sm[t] += sm[t + off];
        __syncthreads();
    }
    if (t == 0) out[0] = sm[0];
}

extern "C" void kernel_launch(void* const* d_in, const int* in_sizes, int n_in,
                              void* d_out, int out_size, void* d_ws, size_t ws_size,
                              hipStream_t stream) {
    const float* scores  = (const float*)d_in[0];
    const int*   werRank = (const int*)d_in[1];
    // d_in[2] is nBest (scalar on device); fixed at 50 per the reference.

    const int B = in_sizes[0] / NBEST;                       // 16384
    const int nblocks = (B + SEGS_PER_BLOCK - 1) / SEGS_PER_BLOCK;  // 2048

    float* partials = (float*)d_ws;                          // 2048 * 4 B = 8 KB

    margin_loss_wmma_kernel<<<nblocks, 32, 0, stream>>>(scores, werRank, partials, B);
    reduce_partials_kernel<<<1, 256, 0, stream>>>(partials, nblocks, (float*)d_out);
}